// MemModule_10539849744492
// MI455X (gfx1250) — compile-verified
//
#include <hip/hip_runtime.h>

typedef __attribute__((ext_vector_type(2))) float v2f;
typedef __attribute__((ext_vector_type(8))) float v8f;

#define B_SZ 256
#define M_SZ 64
#define N_SZ 1024
#define D_SZ 128
#define ND   131072          // N_SZ * D_SZ
#define KSPLIT 64
#define KCHUNK (ND / KSPLIT) // 2048

// ---------------------------------------------------------------------------
// zero the att_node logit accumulator (workspace) each call
// ---------------------------------------------------------------------------
__global__ void zero_ws_kernel(float* __restrict__ p, int n) {
  int i = blockIdx.x * blockDim.x + threadIdx.x;
  if (i < n) p[i] = 0.0f;
}

// ---------------------------------------------------------------------------
// Phase 1: att_node logits, split-K WMMA f32 16x16x4 + atomic accumulate.
// grid = (B/16, KSPLIT), block = 32 (one wave).
// A tile: 16 batch rows x 4 K;  B tile: 4 K x 16 memory slots (m).
// ---------------------------------------------------------------------------
__global__ __launch_bounds__(32)
void att_node_gemm_kernel(const float* __restrict__ node,   // [B, ND]
                          const float* __restrict__ wnode,  // [M, ND]
                          float* __restrict__ acc)          // [B, M] logits
{
  const int  lane = threadIdx.x;
  const int  r15  = lane & 15;            // A row (b) / B col (m) within tile
  const int  klo  = (lane >> 4) << 1;     // K sub-offset: 0 or 2
  const int  b0   = blockIdx.x << 4;
  const long kb   = (long)blockIdx.y * KCHUNK;

  v8f c0 = {}, c1 = {}, c2 = {}, c3 = {};

  const float* an = node  + (long)(b0 + r15) * ND + kb + klo;
  const float* w  = wnode + kb + klo;

  for (int k = 0; k < KCHUNK; k += 4) {
    v2f a  = *(const v2f*)(an + k);
    v2f w0 = *(const v2f*)(w + (long)( 0 + r15) * ND + k);
    v2f w1 = *(const v2f*)(w + (long)(16 + r15) * ND + k);
    v2f w2 = *(const v2f*)(w + (long)(32 + r15) * ND + k);
    v2f w3 = *(const v2f*)(w + (long)(48 + r15) * ND + k);
    c0 = __builtin_amdgcn_wmma_f32_16x16x4_f32(false, a, false, w0, (short)0, c0, false, false);
    c1 = __builtin_amdgcn_wmma_f32_16x16x4_f32(false, a, false, w1, (short)0, c1, false, false);
    c2 = __builtin_amdgcn_wmma_f32_16x16x4_f32(false, a, false, w2, (short)0, c2, false, false);
    c3 = __builtin_amdgcn_wmma_f32_16x16x4_f32(false, a, false, w3, (short)0, c3, false, false);
  }

#pragma unroll
  for (int r = 0; r < 8; ++r) {
    const int bi = b0 + ((lane < 16) ? r : (r + 8));
    atomicAdd(&acc[bi * M_SZ +  0 + r15], c0[r]);
    atomicAdd(&acc[bi * M_SZ + 16 + r15], c1[r]);
    atomicAdd(&acc[bi * M_SZ + 32 + r15], c2[r]);
    atomicAdd(&acc[bi * M_SZ + 48 + r15], c3[r]);
  }
}

// ---------------------------------------------------------------------------
// softmax -> hard_shrink_relu -> l1_normalize over M=64 (one block per batch)
// ---------------------------------------------------------------------------
__device__ __forceinline__ float attn_pipeline(float x, float* sm, int m, float lambd) {
  // max reduction
  sm[m] = x; __syncthreads();
  for (int s = 32; s > 0; s >>= 1) { if (m < s) sm[m] = fmaxf(sm[m], sm[m + s]); __syncthreads(); }
  const float mx = sm[0]; __syncthreads();
  // softmax
  const float e = __expf(x - mx);
  sm[m] = e; __syncthreads();
  for (int s = 32; s > 0; s >>= 1) { if (m < s) sm[m] += sm[m + s]; __syncthreads(); }
  const float p = e / sm[0]; __syncthreads();
  // hard_shrink_relu: relu(p - lambd) * p / (|p - lambd| + eps)
  const float d  = p - lambd;
  const float hs = fmaxf(d, 0.0f) * p / (fabsf(d) + 1e-12f);
  // l1 normalize
  sm[m] = fabsf(hs); __syncthreads();
  for (int s = 32; s > 0; s >>= 1) { if (m < s) sm[m] += sm[m + s]; __syncthreads(); }
  const float denom = fmaxf(sm[0], 1e-12f); __syncthreads();
  return hs / denom;
}

__global__ __launch_bounds__(64)
void att_node_post_kernel(const float* __restrict__ logits, float* __restrict__ att) {
  __shared__ float sm[64];
  const int b = blockIdx.x, m = threadIdx.x;
  const float x = logits[b * M_SZ + m];
  att[b * M_SZ + m] = attn_pipeline(x, sm, m, 0.2f / (float)M_SZ);
}

// ---------------------------------------------------------------------------
// Phase 3: output_node = att_node @ weight_node, WMMA f32 16x16x4, K = M = 64.
// grid = (ND/16, B/16), block = 32 (one wave).
// ---------------------------------------------------------------------------
__global__ __launch_bounds__(32)
void out_node_gemm_kernel(const float* __restrict__ att,    // [B, M]
                          const float* __restrict__ wnode,  // [M, ND]
                          float* __restrict__ out)          // [B, ND]
{
  const int  lane = threadIdx.x;
  const int  r15  = lane & 15;           // A row (b) / B col (nd) within tile
  const int  klo  = (lane >> 4) << 1;    // K (m) sub-offset: 0 or 2
  const long k0   = (long)blockIdx.x << 4;
  const int  b0   = blockIdx.y << 4;

  v8f c = {};
#pragma unroll
  for (int m = 0; m < M_SZ; m += 4) {
    v2f a = *(const v2f*)(att + (b0 + r15) * M_SZ + m + klo);
    v2f bv;
    bv.x = wnode[(long)(m + klo    ) * ND + k0 + r15];
    bv.y = wnode[(long)(m + klo + 1) * ND + k0 + r15];
    c = __builtin_amdgcn_wmma_f32_16x16x4_f32(false, a, false, bv, (short)0, c, false, false);
  }

#pragma unroll
  for (int r = 0; r < 8; ++r) {
    const int bi = b0 + ((lane < 16) ? r : (r + 8));
    out[(long)bi * ND + k0 + r15] = c[r];
  }
}

// ---------------------------------------------------------------------------
// Graph path (tiny): logits + softmax pipeline, then rank-64 expansion.
// ---------------------------------------------------------------------------
__global__ __launch_bounds__(64)
void att_graph_kernel(const float* __restrict__ ge,   // [B, D]
                      const float* __restrict__ wg,   // [Mg, D]
                      float* __restrict__ attg)       // [B, Mg]
{
  __shared__ float sm[64];
  const int b = blockIdx.x, m = threadIdx.x;
  const float* g = ge + b * D_SZ;
  const float* w = wg + m * D_SZ;
  float s = 0.0f;
#pragma unroll 8
  for (int d = 0; d < D_SZ; ++d) s += g[d] * w[d];
  attg[b * M_SZ + m] = attn_pipeline(s, sm, m, 0.2f / (float)M_SZ);
}

__global__ __launch_bounds__(128)
void out_graph_kernel(const float* __restrict__ attg, // [B, Mg]
                      const float* __restrict__ wg,   // [Mg, D]
                      float* __restrict__ outg)       // [B, D]
{
  const int b = blockIdx.x, d = threadIdx.x;
  float s = 0.0f;
#pragma unroll 8
  for (int m = 0; m < M_SZ; ++m) s += attg[b * M_SZ + m] * wg[m * D_SZ + d];
  outg[b * D_SZ + d] = s;
}

// ---------------------------------------------------------------------------
extern "C" void kernel_launch(void* const* d_in, const int* in_sizes, int n_in,
                              void* d_out, int out_size, void* d_ws, size_t ws_size,
                              hipStream_t stream) {
  const float* node = (const float*)d_in[0];   // [256,1024,128]
  const float* ge   = (const float*)d_in[1];   // [256,128]
  const float* wn   = (const float*)d_in[2];   // [64,1024,128]
  const float* wg   = (const float*)d_in[3];   // [64,128]

  float* out_node  = (float*)d_out;                         // 33554432
  float* out_graph = out_node + (long)B_SZ * ND;            // 32768
  float* att_node  = out_graph + B_SZ * D_SZ;               // 16384
  float* att_graph = att_node + B_SZ * M_SZ;                // 16384

  float* ws_logits = (float*)d_ws;                          // [256,64]

  zero_ws_kernel<<<64, 256, 0, stream>>>(ws_logits, B_SZ * M_SZ);
  att_node_gemm_kernel<<<dim3(B_SZ / 16, KSPLIT), 32, 0, stream>>>(node, wn, ws_logits);
  att_node_post_kernel<<<B_SZ, 64, 0, stream>>>(ws_logits, att_node);
  out_node_gemm_kernel<<<dim3(ND / 16, B_SZ / 16), 32, 0, stream>>>(att_node, wn, out_node);
  att_graph_kernel<<<B_SZ, 64, 0, stream>>>(ge, wg, att_graph);
  out_graph_kernel<<<B_SZ, 128, 0, stream>>>(att_graph, wg, out_graph);
}